// MinkowskiConcat_76613626626687
// MI455X (gfx1250) — compile-verified
//
#include <hip/hip_runtime.h>

// MinkowskiConcat forward for MI455X (gfx1250, wave32).
// Pure gather + scatter-add: memory/atomics bound (~2.5 GB traffic, ~0 FLOPs).
// Strategy:
//   Kernel 1: zero the 805 MB output with 128-bit non-temporal stores.
//   Kernel 2: one wave32 per input row; each lane moves one float4:
//             - global_load_b128 (NT) gather of feat[map_in[row]]
//             - 4x global_atomic_add_f32 (no-return, relaxed, agent scope)
//               into out[map_out[row], coloff + lane*4 .. +3]
// Atomics are mandatory: map*_out contains duplicate output rows.

typedef float v4f __attribute__((ext_vector_type(4)));

#define C_IN   128
#define C_OUT  256
#define WAVES_PER_BLOCK 8   // 256 threads / wave32

__global__ void mink_concat_zero_kernel(v4f* __restrict__ out4, long long n4) {
    long long i = (long long)blockIdx.x * blockDim.x + threadIdx.x;
    if (i < n4) {
        v4f z = {0.0f, 0.0f, 0.0f, 0.0f};
        // NT store: use-once streaming fill, don't thrash L2 (out >> 192MB L2 anyway)
        __builtin_nontemporal_store(z, out4 + i);
    }
}

__global__ void mink_concat_scatter_kernel(
    const float* __restrict__ feat1,
    const float* __restrict__ feat2,
    const int*   __restrict__ map1_in,
    const int*   __restrict__ map1_out,
    const int*   __restrict__ map2_in,
    const int*   __restrict__ map2_out,
    float*       __restrict__ out,
    int n1, int n2)
{
    const int wave = threadIdx.x >> 5;          // wave32
    const int lane = threadIdx.x & 31;
    long long row = (long long)blockIdx.x * WAVES_PER_BLOCK + wave;

    const float* src;
    long long in_idx, out_idx;
    int coloff;
    if (row < n1) {
        in_idx  = (long long)map1_in[row];
        out_idx = (long long)map1_out[row];
        src = feat1;
        coloff = 0;
    } else {
        long long r = row - n1;
        if (r >= n2) return;
        in_idx  = (long long)map2_in[r];
        out_idx = (long long)map2_out[r];
        src = feat2;
        coloff = C_IN;
    }

    // Gather one 128-float row: lane l loads float4 l -> one coalesced 512B
    // wave transaction. NT hint: feat rows are (almost always) read once.
    const v4f* srow = (const v4f*)(src + in_idx * C_IN);
    v4f v = __builtin_nontemporal_load(srow + lane);

    // Scatter-add into the output row slice. 4 native f32 atomics per lane;
    // the 4 instructions together cover a contiguous 512B span per wave.
    float* dst = out + out_idx * (long long)C_OUT + coloff + lane * 4;
    __hip_atomic_fetch_add(dst + 0, v.x, __ATOMIC_RELAXED, __HIP_MEMORY_SCOPE_AGENT);
    __hip_atomic_fetch_add(dst + 1, v.y, __ATOMIC_RELAXED, __HIP_MEMORY_SCOPE_AGENT);
    __hip_atomic_fetch_add(dst + 2, v.z, __ATOMIC_RELAXED, __HIP_MEMORY_SCOPE_AGENT);
    __hip_atomic_fetch_add(dst + 3, v.w, __ATOMIC_RELAXED, __HIP_MEMORY_SCOPE_AGENT);
}

extern "C" void kernel_launch(void* const* d_in, const int* in_sizes, int n_in,
                              void* d_out, int out_size, void* d_ws, size_t ws_size,
                              hipStream_t stream) {
    const float* feat1   = (const float*)d_in[0];
    const float* feat2   = (const float*)d_in[1];
    const int*   map1_in  = (const int*)d_in[2];
    const int*   map1_out = (const int*)d_in[3];
    const int*   map2_in  = (const int*)d_in[4];
    const int*   map2_out = (const int*)d_in[5];
    float* out = (float*)d_out;

    const int n1 = in_sizes[2];   // number of feat1 rows being scattered
    const int n2 = in_sizes[4];   // number of feat2 rows being scattered

    // --- Kernel 1: zero-fill output (out_size f32 elements) ---
    long long n4 = (long long)out_size / 4;          // float4 count
    int zthreads = 256;
    long long zblocks = (n4 + zthreads - 1) / zthreads;
    mink_concat_zero_kernel<<<dim3((unsigned)zblocks), dim3(zthreads), 0, stream>>>(
        (v4f*)out, n4);

    // --- Kernel 2: gather + atomic scatter-add, one wave32 per row ---
    long long nwaves = (long long)n1 + (long long)n2;
    long long blocks = (nwaves + WAVES_PER_BLOCK - 1) / WAVES_PER_BLOCK;
    mink_concat_scatter_kernel<<<dim3((unsigned)blocks), dim3(32 * WAVES_PER_BLOCK), 0, stream>>>(
        feat1, feat2, map1_in, map1_out, map2_in, map2_out, out, n1, n2);
}